// VisionDynamicSparseAttention_3298534884133
// MI455X (gfx1250) — compile-verified
//
#include <hip/hip_runtime.h>

// Problem constants (match reference setup_inputs)
#define NH    12
#define HD    64
#define SEQ   2048
#define HID   768
#define KTOK  1024          // int(S * 0.5)
#define SCALE 0.125f        // 64^-0.5

typedef __attribute__((ext_vector_type(2))) float v2f;
typedef __attribute__((ext_vector_type(8))) float v8f;

// D = A(16x4 f32) * B(4x16 f32) + C, exact fp32 via CDNA5 WMMA
static __device__ __forceinline__ v8f wmma4(v2f a, v2f b, v8f c) {
  return __builtin_amdgcn_wmma_f32_16x16x4_f32(
      /*neg_a=*/false, a, /*neg_b=*/false, b,
      /*c_mod=*/(short)0, c, /*reuse_a=*/false, /*reuse_b=*/false);
}

// monotone uint key for fp32 ordering (exact involution pair, no NaNs here)
static __device__ __forceinline__ unsigned fkey(float f) {
  unsigned u = __float_as_uint(f);
  return (u & 0x80000000u) ? ~u : (u | 0x80000000u);
}
static __device__ __forceinline__ float ikey(unsigned u) {
  unsigned b = (u & 0x80000000u) ? (u & 0x7FFFFFFFu) : ~u;
  return __uint_as_float(b);
}

// ---------------------------------------------------------------------------
// Kernel 1: qkv = X @ W^T + b, scattered to Q/K/V in [B, nh, S, hd]
// Block = 128 threads (4 waves). Each wave computes a 32x32 tile
// (4 WMMA accumulators, A/B fragments reused 2x each -> 4 loads : 4 wmma).
// ---------------------------------------------------------------------------
__global__ __launch_bounds__(128)
void qkv_gemm(const float* __restrict__ X, const float* __restrict__ W,
              const float* __restrict__ bias,
              float* __restrict__ Q, float* __restrict__ Kd,
              float* __restrict__ V) {
  const int lane = threadIdx.x & 31;
  const int wave = threadIdx.x >> 5;
  const int m0   = blockIdx.x * 64 + (wave & 1) * 32;
  const int n0   = blockIdx.y * 64 + (wave >> 1) * 32;
  const int half = lane >> 4;     // 0: K pair {0,1}, 1: K pair {2,3}
  const int l15  = lane & 15;
  const int ka   = half * 2;

  const float* A0 = X + (size_t)(m0 + l15) * HID;
  const float* A1 = A0 + (size_t)16 * HID;
  const float* B0 = W + (size_t)(n0 + l15) * HID;
  const float* B1 = B0 + (size_t)16 * HID;

  v8f acc00 = {}, acc01 = {}, acc10 = {}, acc11 = {};
#pragma unroll 4
  for (int k0 = 0; k0 < HID; k0 += 4) {
    v2f a0 = *(const v2f*)(A0 + k0 + ka);
    v2f a1 = *(const v2f*)(A1 + k0 + ka);
    v2f b0 = *(const v2f*)(B0 + k0 + ka);
    v2f b1 = *(const v2f*)(B1 + k0 + ka);
    acc00 = wmma4(a0, b0, acc00);
    acc01 = wmma4(a0, b1, acc01);
    acc10 = wmma4(a1, b0, acc10);
    acc11 = wmma4(a1, b1, acc11);
  }

#pragma unroll
  for (int tm = 0; tm < 2; ++tm) {
#pragma unroll
    for (int tn = 0; tn < 2; ++tn) {
      const v8f acc = (tm == 0) ? ((tn == 0) ? acc00 : acc01)
                                : ((tn == 0) ? acc10 : acc11);
      const int n    = n0 + tn * 16 + l15;
      const int sec  = n / HID;            // 0=q 1=k 2=v
      const int hcol = n - sec * HID;
      const int head = hcol >> 6;
      const int d    = hcol & 63;
      float* dst = (sec == 0) ? Q : (sec == 1) ? Kd : V;
      const float bn = bias[n];
#pragma unroll
      for (int r = 0; r < 8; ++r) {
        int m  = m0 + tm * 16 + r + half * 8;  // lanes16-31 hold M=r+8
        int b_ = m >> 11;                      // / SEQ
        int s  = m & (SEQ - 1);
        dst[(((size_t)b_ * NH + head) * SEQ + s) * HD + d] = acc[r] + bn;
      }
    }
  }
}

// ---------------------------------------------------------------------------
// Kernel 2: fused attention for one (b,h) x 16-query tile per block.
// 256 threads = 8 waves. Scores (16 x 2048 fp32 = 128 KB) live in LDS.
// Phase 1: QK^T via WMMA (Q fragment hoisted to VGPRs).
// Phase 2: exact k-th-largest threshold + softmax; row keys cached in VGPRs
//          so the 32-bit search is pure VALU (one LDS read + write pass).
// Phase 3: P @ V with split-K over waves, LDS combine, write ctx [B,S,H].
// ---------------------------------------------------------------------------
__global__ __launch_bounds__(256)
void attn_fused(const float* __restrict__ Q, const float* __restrict__ Kd,
                const float* __restrict__ V, float* __restrict__ ctx) {
  __shared__ float sc[16 * SEQ];     // score rows / prob rows
  __shared__ float part[16 * 64];    // split-K partials

  const int lane = threadIdx.x & 31;
  const int wave = threadIdx.x >> 5;
  const int bh   = blockIdx.x >> 7;              // (b*NH + h)
  const int q0   = (blockIdx.x & 127) * 16;
  const int half = lane >> 4;
  const int l15  = lane & 15;
  const int ka   = half * 2;

  // ---- Phase 1: scores = scale * Q_tile @ K^T, into LDS ----
  {
    const float* qbase = Q + ((size_t)bh * SEQ + q0 + l15) * HD;
    v2f qa[16];
#pragma unroll
    for (int t = 0; t < 16; ++t)
      qa[t] = *(const v2f*)(qbase + t * 4 + ka);

    for (int kt = wave * 16; kt < SEQ; kt += 128) {
      const float* kbase = Kd + ((size_t)bh * SEQ + kt + l15) * HD;
      v8f acc = {};
#pragma unroll
      for (int t = 0; t < 16; ++t) {
        v2f b = *(const v2f*)(kbase + t * 4 + ka);
        acc = wmma4(qa[t], b, acc);
      }
#pragma unroll
      for (int r = 0; r < 8; ++r)
        sc[(r + half * 8) * SEQ + kt + l15] = acc[r] * SCALE;
    }
  }
  __syncthreads();

  // ---- Phase 2: per-row exact top-k threshold + softmax (wave owns 2 rows)
  for (int rr = 0; rr < 2; ++rr) {
    float* row = sc + (wave + rr * 8) * SEQ;

    // one LDS read pass: cache monotone keys in registers
    unsigned keys[64];
    unsigned mk = 0u;
#pragma unroll
    for (int t = 0; t < 64; ++t) {
      keys[t] = fkey(row[lane + t * 32]);
      mk = (keys[t] > mk) ? keys[t] : mk;
    }
    for (int off = 16; off > 0; off >>= 1) {
      unsigned o = __shfl_down(mk, off);
      mk = (o > mk) ? o : mk;
    }
    mk = __shfl(mk, 0);
    const float mx = ikey(mk);

    // k-th largest key: max u with count(keys >= u) >= KTOK (bitwise search)
    unsigned cand = 0u;
    for (int bit = 31; bit >= 0; --bit) {
      const unsigned trial = cand | (1u << bit);
      int c = 0;
#pragma unroll
      for (int t = 0; t < 64; ++t) c += (keys[t] >= trial) ? 1 : 0;
      for (int off = 16; off > 0; off >>= 1) c += __shfl_down(c, off);
      c = __shfl(c, 0);
      if (c >= KTOK) cand = trial;
    }

    // exp pass (dropped entries -> exp(NEG_INF)=0), numerators to LDS
    float sum = 0.0f;
#pragma unroll
    for (int t = 0; t < 64; ++t) {
      float e = (keys[t] >= cand) ? __expf(ikey(keys[t]) - mx) : 0.0f;
      row[lane + t * 32] = e;
      sum += e;
    }
    for (int off = 16; off > 0; off >>= 1) sum += __shfl_down(sum, off);
    sum = __shfl(sum, 0);
    const float rinv = 1.0f / sum;
#pragma unroll
    for (int t = 0; t < 64; ++t) row[lane + t * 32] *= rinv;
  }
  __syncthreads();

  // ---- Phase 3: out16x64 = P(16x2048) @ V(2048x64), split-K over 2 halves
  const int nsub = (wave & 3) * 16;        // which 16 cols of hd
  const int kh   = (wave >> 2) * (SEQ / 2);
  const float* vbase = V + (size_t)bh * SEQ * HD;
  v8f acc = {};
  for (int k0 = kh; k0 < kh + SEQ / 2; k0 += 4) {
    v2f a = *(const v2f*)(sc + l15 * SEQ + k0 + ka);   // P[m][k..k+1] from LDS
    v2f b;
    b.x = vbase[(size_t)(k0 + ka) * HD + nsub + l15];  // V[k][n]
    b.y = vbase[(size_t)(k0 + ka + 1) * HD + nsub + l15];
    acc = wmma4(a, b, acc);
  }
  if (wave >= 4) {
#pragma unroll
    for (int r = 0; r < 8; ++r)
      part[(r + half * 8) * 64 + nsub + l15] = acc[r];
  }
  __syncthreads();
  if (wave < 4) {
    const int b_ = bh / NH, h_ = bh - b_ * NH;
#pragma unroll
    for (int r = 0; r < 8; ++r) {
      int m = r + half * 8;
      float val = acc[r] + part[m * 64 + nsub + l15];
      ctx[((size_t)b_ * SEQ + q0 + m) * HID + h_ * HD + nsub + l15] = val;
    }
  }
}

// ---------------------------------------------------------------------------
// Kernel 3: out = ctx @ proj_w^T + proj_b   (M=B*S, N=K=768)
// Same 32x32-per-wave structure as kernel 1.
// ---------------------------------------------------------------------------
__global__ __launch_bounds__(128)
void proj_gemm(const float* __restrict__ Xc, const float* __restrict__ W,
               const float* __restrict__ bias, float* __restrict__ out) {
  const int lane = threadIdx.x & 31;
  const int wave = threadIdx.x >> 5;
  const int m0   = blockIdx.x * 64 + (wave & 1) * 32;
  const int n0   = blockIdx.y * 64 + (wave >> 1) * 32;
  const int half = lane >> 4;
  const int l15  = lane & 15;
  const int ka   = half * 2;

  const float* A0 = Xc + (size_t)(m0 + l15) * HID;
  const float* A1 = A0 + (size_t)16 * HID;
  const float* B0 = W + (size_t)(n0 + l15) * HID;
  const float* B1 = B0 + (size_t)16 * HID;

  v8f acc00 = {}, acc01 = {}, acc10 = {}, acc11 = {};
#pragma unroll 4
  for (int k0 = 0; k0 < HID; k0 += 4) {
    v2f a0 = *(const v2f*)(A0 + k0 + ka);
    v2f a1 = *(const v2f*)(A1 + k0 + ka);
    v2f b0 = *(const v2f*)(B0 + k0 + ka);
    v2f b1 = *(const v2f*)(B1 + k0 + ka);
    acc00 = wmma4(a0, b0, acc00);
    acc01 = wmma4(a0, b1, acc01);
    acc10 = wmma4(a1, b0, acc10);
    acc11 = wmma4(a1, b1, acc11);
  }

#pragma unroll
  for (int tm = 0; tm < 2; ++tm) {
#pragma unroll
    for (int tn = 0; tn < 2; ++tn) {
      const v8f acc = (tm == 0) ? ((tn == 0) ? acc00 : acc01)
                                : ((tn == 0) ? acc10 : acc11);
      const int n = n0 + tn * 16 + l15;
      const float bn = bias[n];
#pragma unroll
      for (int r = 0; r < 8; ++r) {
        int m = m0 + tm * 16 + r + half * 8;
        out[(size_t)m * HID + n] = acc[r] + bn;
      }
    }
  }
}

extern "C" void kernel_launch(void* const* d_in, const int* in_sizes, int n_in,
                              void* d_out, int out_size, void* d_ws, size_t ws_size,
                              hipStream_t stream) {
  const float* hs     = (const float*)d_in[0];
  const float* qkv_w  = (const float*)d_in[1];
  const float* qkv_b  = (const float*)d_in[2];
  const float* proj_w = (const float*)d_in[3];
  const float* proj_b = (const float*)d_in[4];
  float* out = (float*)d_out;

  const int B = in_sizes[0] / (SEQ * HID);      // = 2
  const size_t per = (size_t)B * NH * SEQ * HD; // one of q/k/v

  float* Q   = (float*)d_ws;
  float* Kd  = Q + per;
  float* V   = Kd + per;
  float* ctx = V + per;                          // [B*S, HID]

  dim3 g1((B * SEQ) / 64, (3 * HID) / 64);
  qkv_gemm<<<g1, 128, 0, stream>>>(hs, qkv_w, qkv_b, Q, Kd, V);

  attn_fused<<<B * NH * (SEQ / 16), 256, 0, stream>>>(Q, Kd, V, ctx);

  dim3 g3((B * SEQ) / 64, HID / 64);
  proj_gemm<<<g3, 128, 0, stream>>>(ctx, proj_w, proj_b, out);
}